// LindBladEvolve_79508434583799
// MI455X (gfx1250) — compile-verified
//
#include <hip/hip_runtime.h>
#include <hip/hip_bf16.h>
#include <stdint.h>

// ---------------- problem constants ----------------
#define DT_F   0.01f
#define T1     4095            // T_STEPS - 1
#define NTRAJ  2048
#define NCTRL  2
#define NCOP   2
#define TABW   12              // floats per step: H00r,H11r,H01r,H01i, P (4 cplx)
#define CHUNK  512             // steps staged into LDS per TDM transfer
#define NCHUNK ((T1 + CHUNK - 1) / CHUNK)   // 8 (last chunk = 511 steps)

// output layout (floats), concatenated in reference return order
__device__ constexpr long OFF_PROB   = (long)NTRAJ * 4;                 // 8192
__device__ constexpr long OFF_EN     = OFF_PROB + NTRAJ;                // 10240
__device__ constexpr long OFF_EV     = OFF_EN + NTRAJ;                  // 12288
__device__ constexpr long OFF_STATES = OFF_EV + (long)T1 * NTRAJ;       // 8398848

// ---------------- complex helpers ----------------
struct cplx { float x, y; };
__device__ __forceinline__ cplx mk(float a, float b) { cplx c; c.x = a; c.y = b; return c; }
__device__ __forceinline__ cplx operator+(cplx a, cplx b) { return mk(a.x + b.x, a.y + b.y); }
__device__ __forceinline__ cplx operator-(cplx a, cplx b) { return mk(a.x - b.x, a.y - b.y); }
__device__ __forceinline__ cplx operator*(cplx a, cplx b) { return mk(a.x * b.x - a.y * b.y, a.x * b.y + a.y * b.x); }
__device__ __forceinline__ cplx operator*(float s, cplx a) { return mk(s * a.x, s * a.y); }
__device__ __forceinline__ cplx cconj(cplx a) { return mk(a.x, -a.y); }
__device__ __forceinline__ cplx cdiv(cplx a, cplx b) {
  float n = b.x * b.x + b.y * b.y;
  return mk((a.x * b.x + a.y * b.y) / n, (a.y * b.x - a.x * b.y) / n);
}
__device__ __forceinline__ cplx cexp_(cplx z) {
  float e = expf(z.x);
  return mk(e * cosf(z.y), e * sinf(z.y));
}
__device__ __forceinline__ cplx csqrt_(cplx z) {   // principal branch
  float m  = hypotf(z.x, z.y);
  float re = sqrtf(fmaxf(0.5f * (m + z.x), 0.0f));
  float im = sqrtf(fmaxf(0.5f * (m - z.x), 0.0f));
  return mk(re, copysignf(im, z.y));
}

// ---------------- kernel 1: per-step uniform table ----------------
// tab[t*12 + {0..3}]  = H00.re, H11.re, H01.re, H01.im   (H_t is Hermitian)
// tab[t*12 + {4..11}] = P00, P01, P10, P11 (re,im interleaved), P = expm(-i*dt*Heff)
__global__ void __launch_bounds__(256) precompute_kernel(
    const float* __restrict__ ham0_re, const float* __restrict__ ham0_im,
    const float* __restrict__ hamc_re, const float* __restrict__ hamc_im,
    const float* __restrict__ cops_re, const float* __restrict__ cops_im,
    const float* __restrict__ ctrls_real, const float* __restrict__ ctrls_im,
    float* __restrict__ tab)
{
  int t = blockIdx.x * blockDim.x + threadIdx.x;
  if (t >= T1) return;

  cplx H[2][2];
  {
    cplx M0[2][2];
    for (int i = 0; i < 2; ++i)
      for (int j = 0; j < 2; ++j)
        M0[i][j] = mk(ham0_re[i * 2 + j], ham0_im[i * 2 + j]);
    for (int i = 0; i < 2; ++i)
      for (int j = 0; j < 2; ++j)
        H[i][j] = M0[i][j] + cconj(M0[j][i]);
  }
  for (int jc = 0; jc < NCTRL; ++jc) {
    float cr = ctrls_real[t * NCTRL + jc];
    float ci = ctrls_im[t * NCTRL + jc];
    cplx Mc[2][2];
    for (int i = 0; i < 2; ++i)
      for (int j = 0; j < 2; ++j)
        Mc[i][j] = mk(hamc_re[jc * 4 + i * 2 + j], hamc_im[jc * 4 + i * 2 + j]);
    for (int i = 0; i < 2; ++i)
      for (int j = 0; j < 2; ++j) {
        cplx apb = Mc[i][j] + cconj(Mc[j][i]);   // A+B (Hermitian)
        cplx amb = Mc[i][j] - cconj(Mc[j][i]);   // A-B (anti-Hermitian)
        // H += cr*apb + i*ci*amb
        H[i][j].x += cr * apb.x - ci * amb.y;
        H[i][j].y += cr * apb.y + ci * amb.x;
      }
  }
  // sum_k c_k^dag c_k
  cplx S[2][2] = {{mk(0, 0), mk(0, 0)}, {mk(0, 0), mk(0, 0)}};
  for (int k = 0; k < NCOP; ++k) {
    cplx c[2][2];
    for (int i = 0; i < 2; ++i)
      for (int j = 0; j < 2; ++j)
        c[i][j] = mk(cops_re[k * 4 + i * 2 + j], cops_im[k * 4 + i * 2 + j]);
    for (int a = 0; a < 2; ++a)
      for (int b = 0; b < 2; ++b)
        S[a][b] = S[a][b] + cconj(c[0][a]) * c[0][b] + cconj(c[1][a]) * c[1][b];
  }
  // M = -i*DT*(H - 0.5i*S)
  cplx Mm[2][2];
  for (int a = 0; a < 2; ++a)
    for (int b = 0; b < 2; ++b) {
      cplx heff = mk(H[a][b].x + 0.5f * S[a][b].y, H[a][b].y - 0.5f * S[a][b].x);
      Mm[a][b]  = mk(DT_F * heff.y, -DT_F * heff.x);   // -i * heff * DT
    }
  // closed-form expm 2x2
  cplx mu  = mk(0.5f * (Mm[0][0].x + Mm[1][1].x), 0.5f * (Mm[0][0].y + Mm[1][1].y));
  cplx det = Mm[0][0] * Mm[1][1] - Mm[0][1] * Mm[1][0];
  cplx d2  = mu * mu - det;
  cplx d   = csqrt_(d2);
  float ad = hypotf(d.x, d.y);
  cplx ch, sh;
  {
    float chx = coshf(d.x), shx = sinhf(d.x), cy = cosf(d.y), sy = sinf(d.y);
    ch = mk(chx * cy, shx * sy);
    sh = mk(shx * cy, chx * sy);
  }
  cplx ratio = (ad < 1e-6f) ? mk(1.0f + d2.x * (1.0f / 6.0f), d2.y * (1.0f / 6.0f))
                            : cdiv(sh, d);
  cplx emu = cexp_(mu);
  cplx P[2][2];
  for (int a = 0; a < 2; ++a)
    for (int b = 0; b < 2; ++b) {
      cplx base = ratio * ((a == b) ? (Mm[a][b] - mu) : Mm[a][b]);
      if (a == b) base = base + ch;
      P[a][b] = emu * base;
    }
  float* o = tab + (long)t * TABW;
  o[0] = H[0][0].x; o[1] = H[1][1].x; o[2] = H[0][1].x; o[3] = H[0][1].y;
  o[4] = P[0][0].x; o[5] = P[0][0].y; o[6]  = P[0][1].x; o[7]  = P[0][1].y;
  o[8] = P[1][0].x; o[9] = P[1][0].y; o[10] = P[1][1].x; o[11] = P[1][1].y;
}

// ---------------- TDM staging (CDNA5 Tensor Data Mover) ----------------
#if defined(__has_builtin)
#if __has_builtin(__builtin_amdgcn_tensor_load_to_lds)
#define HAVE_TDM 1
#endif
#endif

#ifdef HAVE_TDM
typedef unsigned int u32x4 __attribute__((ext_vector_type(4)));
typedef int          i32x4 __attribute__((ext_vector_type(4)));
typedef int          i32x8 __attribute__((ext_vector_type(8)));

// One 1-D TDM transfer: ndw dwords, global -> LDS (per CDNA5 ISA D# layout).
__device__ __forceinline__ void tdm_load(const float* gsrc, const float* ldst, unsigned ndw)
{
  unsigned long long ga = (unsigned long long)(uintptr_t)gsrc;
  unsigned int la = (unsigned int)(uintptr_t)ldst;   // low 32 bits of flat addr = LDS offset
  // Group0: count=1 | lds_addr | global_addr[31:0] | ga[56:32] + type=2 in [127:126]
  u32x4 g0 = { 1u, la, (unsigned int)ga,
               ((unsigned int)((ga >> 32) & 0x01FFFFFFull)) | 0x80000000u };
  // Group1: data_size=4B; tensor_dim0 = tile_dim0 = stride0 = ndw; tensor_dim1 = 1
  i32x8 g1 = { (int)0x20000u,          // workgroup_mask=0, data_size=2 (4B)
               (int)(ndw << 16),       // abar_addr=0 | tensor_dim0[15:0]
               (int)(1u << 16),        // tensor_dim0[31:16]=0 | tensor_dim1[15:0]=1
               (int)(ndw << 16),       // tensor_dim1[31:16]=0 | tile_dim0
               0,                      // tile_dim1=0 (unused), tile_dim2=0
               (int)ndw,               // tensor_dim0_stride[31:0]
               0, 0 };                 // stride0[47:32], stride1 = 0
  i32x4 g2 = { 1, 1, 0, 0 };           // tensor_dim2=1, tensor_dim3=1 (benign)
  i32x4 g3 = { 0, (int)(1u << 16), 0, 0 };   // tensor_dim4=1
#if __clang_major__ >= 23
  i32x8 gz = { 0, 0, 0, 0, 0, 0, 0, 0 };
  __builtin_amdgcn_tensor_load_to_lds(g0, g1, g2, g3, gz, 0);
#else
  __builtin_amdgcn_tensor_load_to_lds(g0, g1, g2, g3, 0);
#endif
}
#endif

// ---------------- kernel 2: trajectory scan ----------------
__global__ void __launch_bounds__(256) scan_kernel(
    const float* __restrict__ tab,
    const float* __restrict__ cops_re, const float* __restrict__ cops_im,
    const float* __restrict__ rand_r0,
    const float* __restrict__ rand_r, const float* __restrict__ rand_jump,
    const int* __restrict__ init_idx,
    float* __restrict__ out)
{
  __shared__ __align__(16) float buf[2][CHUNK * TABW];   // 2 x 24 KB double buffer

  int b = blockIdx.x * blockDim.x + threadIdx.x;

  // collapse operators + Hermitian quadratic forms of c^dag c (uniform, L2/scalar cached)
  cplx C[NCOP][2][2];
  float A00[NCOP], A11[NCOP];
  cplx A01[NCOP];
  for (int k = 0; k < NCOP; ++k) {
    for (int i = 0; i < 2; ++i)
      for (int j = 0; j < 2; ++j)
        C[k][i][j] = mk(cops_re[k * 4 + i * 2 + j], cops_im[k * 4 + i * 2 + j]);
    A00[k] = C[k][0][0].x * C[k][0][0].x + C[k][0][0].y * C[k][0][0].y
           + C[k][1][0].x * C[k][1][0].x + C[k][1][0].y * C[k][1][0].y;
    A11[k] = C[k][0][1].x * C[k][0][1].x + C[k][0][1].y * C[k][0][1].y
           + C[k][1][1].x * C[k][1][1].x + C[k][1][1].y * C[k][1][1].y;
    A01[k] = cconj(C[k][0][0]) * C[k][0][1] + cconj(C[k][1][0]) * C[k][1][1];
  }

  int i0 = init_idx[b];
  cplx s0 = mk(i0 == 0 ? 1.0f : 0.0f, 0.0f);
  cplx s1 = mk(i0 == 1 ? 1.0f : 0.0f, 0.0f);
  float r = rand_r0[b], logp = 0.0f, energy = 0.0f;

  float4* out_sts = (float4*)(out + OFF_STATES);

#ifdef HAVE_TDM
  if (threadIdx.x < 32) {                               // wave 0 drives the TDM
    int st0 = (T1 < CHUNK) ? T1 : CHUNK;
    tdm_load(tab, buf[0], (unsigned)(st0 * TABW));
  }
#endif

  for (int c = 0; c < NCHUNK; ++c) {
    int tbase = c * CHUNK;
    int steps = ((T1 - tbase) < CHUNK) ? (T1 - tbase) : CHUNK;

#ifdef HAVE_TDM
    __syncthreads();   // buffer (c+1)&1 no longer being read (chunk c-1 done)
    if (threadIdx.x < 32) {
      if (c + 1 < NCHUNK) {
        int nb = (c + 1) * CHUNK;
        int ns = ((T1 - nb) < CHUNK) ? (T1 - nb) : CHUNK;
        tdm_load(tab + (long)nb * TABW, buf[(c + 1) & 1], (unsigned)(ns * TABW));
        __builtin_amdgcn_s_wait_tensorcnt(1);   // in-order: chunk c complete
      } else {
        __builtin_amdgcn_s_wait_tensorcnt(0);
      }
    }
    __syncthreads();   // chunk c visible to all waves
#else
    __syncthreads();
    {
      const float* src = tab + (long)tbase * TABW;
      int nd = steps * TABW;
      for (int i = threadIdx.x; i < nd; i += (int)blockDim.x)
        buf[c & 1][i] = src[i];
    }
    __syncthreads();
#endif

    const float4* tv = (const float4*)buf[c & 1];
    for (int tt = 0; tt < steps; ++tt) {
      int t = tbase + tt;
      float4 Ah = tv[tt * 3 + 0];                 // H00r, H11r, H01r, H01i
      float4 Pa = tv[tt * 3 + 1];                 // P00, P01
      float4 Pb = tv[tt * 3 + 2];                 // P10, P11
      cplx P00 = mk(Pa.x, Pa.y), P01 = mk(Pa.z, Pa.w);
      cplx P10 = mk(Pb.x, Pb.y), P11 = mk(Pb.z, Pb.w);

      cplx c0 = P00 * s0 + P01 * s1;              // no-jump candidate
      cplx c1 = P10 * s0 + P11 * s1;
      float n2c = c0.x * c0.x + c0.y * c0.y + c1.x * c1.x + c1.y * c1.y;
      bool jump = (n2c <= r) || (n2c < 1e-9f);

      long o = (long)t * NTRAJ + b;
      int ev = 0;
      if (jump) {                                 // rare: whole wave usually skips
        float rr = rand_r[o];
        float ru = rand_jump[o];
        float n0 = s0.x * s0.x + s0.y * s0.y;
        float n1 = s1.x * s1.x + s1.y * s1.y;
        float crr = s0.x * s1.x + s0.y * s1.y;    // conj(s0)*s1
        float cii = s0.x * s1.y - s0.y * s1.x;
        float e0 = A00[0] * n0 + A11[0] * n1 + 2.0f * (A01[0].x * crr - A01[0].y * cii);
        float e1 = A00[1] * n0 + A11[1] * n1 + 2.0f * (A01[1].x * crr - A01[1].y * cii);
        float esum = e0 + e1;
        float jp0 = e0 / esum, jp1 = e1 / esum;
        int idx = (jp0 < ru) ? 1 : 0;
        cplx k00 = idx ? C[1][0][0] : C[0][0][0];
        cplx k01 = idx ? C[1][0][1] : C[0][0][1];
        cplx k10 = idx ? C[1][1][0] : C[0][1][0];
        cplx k11 = idx ? C[1][1][1] : C[0][1][1];
        cplx j0 = k00 * s0 + k01 * s1;
        cplx j1 = k10 * s0 + k11 * s1;
        float jn2 = j0.x * j0.x + j0.y * j0.y + j1.x * j1.x + j1.y * j1.y;
        float sq = sqrtf(jn2);
        float psel = idx ? jp1 : jp0;
        logp += logf(n0 + n1) + logf(psel);
        s0 = mk(j0.x / sq, j0.y / sq);
        s1 = mk(j1.x / sq, j1.y / sq);
        r = rr;
        ev = idx + 1;
      } else {
        s0 = c0;
        s1 = c1;
      }

      // energy with new (unnormalized) state, Hermitian H_t
      float n0 = s0.x * s0.x + s0.y * s0.y;
      float n1 = s1.x * s1.x + s1.y * s1.y;
      float crr = s0.x * s1.x + s0.y * s1.y;
      float cii = s0.x * s1.y - s0.y * s1.x;
      energy += Ah.x * n0 + Ah.y * n1 + 2.0f * (Ah.z * crr - Ah.w * cii);

      out[OFF_EV + o] = (float)ev;
      out_sts[o] = make_float4(s0.x, s0.y, s1.x, s1.y);
    }
  }

  ((float4*)out)[b] = make_float4(s0.x, s0.y, s1.x, s1.y);
  float n2f = s0.x * s0.x + s0.y * s0.y + s1.x * s1.x + s1.y * s1.y;
  out[OFF_PROB + b] = expf(logp) * n2f;
  out[OFF_EN + b]   = energy;
}

// ---------------- launch ----------------
extern "C" void kernel_launch(void* const* d_in, const int* in_sizes, int n_in,
                              void* d_out, int out_size, void* d_ws, size_t ws_size,
                              hipStream_t stream)
{
  const float* ham0_re    = (const float*)d_in[0];
  const float* ham0_im    = (const float*)d_in[1];
  const float* hamc_re    = (const float*)d_in[2];
  const float* hamc_im    = (const float*)d_in[3];
  const float* cops_re    = (const float*)d_in[4];
  const float* cops_im    = (const float*)d_in[5];
  const float* ctrls_real = (const float*)d_in[6];
  const float* ctrls_im   = (const float*)d_in[7];
  const float* rand_r0    = (const float*)d_in[8];
  const float* rand_r     = (const float*)d_in[9];
  const float* rand_jump  = (const float*)d_in[10];
  const int*   init_idx   = (const int*)d_in[11];
  float* out = (float*)d_out;
  float* tab = (float*)d_ws;     // T1*12 floats = 196,560 B

  precompute_kernel<<<(T1 + 255) / 256, 256, 0, stream>>>(
      ham0_re, ham0_im, hamc_re, hamc_im, cops_re, cops_im,
      ctrls_real, ctrls_im, tab);

  scan_kernel<<<NTRAJ / 256, 256, 0, stream>>>(
      tab, cops_re, cops_im, rand_r0, rand_r, rand_jump, init_idx, out);
}